// SoftMaxPlus_6511170421076
// MI455X (gfx1250) — compile-verified
//
#include <hip/hip_runtime.h>

// Prefix-scan of softplus(beta*c)/beta along rows (B=4096, S=8192, f32).
// Memory-bound: 256 MiB traffic -> ~11us at 23.3 TB/s. One wave per row.
// Scan-by-matmul on the WMMA pipe:
//   tile X (16x16, row-major chunk of 256 elems):
//     D1 = X * U            (U = upper-tri ones, incl diag)  -> prefix within rows
//     D  = D1 + Lstrict * Z (Z[k][*] = rowsum_k, built from D1 col 15 via ds_swizzle)
//     out = D + carry;  carry = readlane(out[15][15])
// All f32 via V_WMMA_F32_16X16X4_F32 (4 K-slices per 16x16x16 matmul).
//
// Softplus computed in hardware base-2 (raw v_exp_f32 / v_log_f32, no libm
// range fixups -- inputs are provably safe: exp2 arg <= 0, log2 arg in [1,2]):
//   softplus(b*x)/b = relu(x) + (ln2/b) * log2(1 + exp2(-|(b*log2e)*x|))
// relu(x) as v_med3_f32(x, 0, +inf): single op, no canonicalize.

typedef __attribute__((ext_vector_type(2))) float v2f;
typedef __attribute__((ext_vector_type(8))) float v8f;

// ds_swizzle group-of-32 patterns: src = ((lane & and) | or) ^ xor
//  BCAST: lanes0-15 <- lane15, lanes16-31 <- lane31   (and=0x10, or=0x0F)
#define SWZ_BCAST 0x01F0
//  CROSS: lanes0-15 <- lane31, lanes16-31 <- lane15   (xor=0x10 on top of BCAST)
#define SWZ_CROSS 0x41F0

template <int IMM>
__device__ __forceinline__ float swz(float v) {
  return __int_as_float(__builtin_amdgcn_ds_swizzle(__float_as_int(v), IMM));
}

__device__ __forceinline__ v8f wmma4(v2f a, v2f b, v8f c) {
  // (neg_a, A, neg_b, B, c_mod, C, reuse_a, reuse_b)
  return __builtin_amdgcn_wmma_f32_16x16x4_f32(false, a, false, b, (short)0, c,
                                               false, false);
}

// bl = beta * log2(e), c = ln(2) / beta  (beta > 0)
__device__ __forceinline__ float softplus_beta(float x, float bl, float c) {
  float u = bl * x;
  float e = __builtin_amdgcn_exp2f(-__builtin_fabsf(u));  // v_exp_f32, -|.| free
  float l = __builtin_amdgcn_logf(1.0f + e);              // v_log_f32 (log2)
  float relu = __builtin_amdgcn_fmed3f(x, 0.0f, __builtin_inff());  // v_med3_f32
  return __builtin_fmaf(l, c, relu);
}

__global__ __launch_bounds__(256) void softplus_cumsum_wmma(
    const float* __restrict__ c, const float* __restrict__ beta_p,
    float* __restrict__ out, int B, int S) {
  const int lane = threadIdx.x & 31;
  const int wave = threadIdx.x >> 5;
  const int row = blockIdx.x * (blockDim.x >> 5) + wave;
  if (row >= B) return;

  const float beta = beta_p[0];
  const float bl = beta * 1.4426950408889634f;       // beta * log2(e)
  const float lc = 0.6931471805599453f / beta;       // ln2 / beta

  const int r  = lane & 15;      // M index this lane holds in A / N index in B,C,D
  const int hi = lane >> 4;      // lane-half
  const int c0 = hi << 1;        // K offset (+2) carried by the upper half

  // Constant triangular operands, per-lane (hoisted out of the tile loop).
  v2f UB[4];  // U (upper-tri incl diag) as B operand slices
  v2f LA[4];  // Lstrict as A operand slices
#pragma unroll
  for (int s = 0; s < 4; ++s) {
    int k0 = 4 * s + c0;
    int k1 = k0 + 1;
    UB[s][0] = (r >= k0) ? 1.0f : 0.0f;   // U[k0][r]
    UB[s][1] = (r >= k1) ? 1.0f : 0.0f;   // U[k1][r]
    LA[s][0] = (r > k0) ? 1.0f : 0.0f;    // Lstrict[r][k0]
    LA[s][1] = (r > k1) ? 1.0f : 0.0f;    // Lstrict[r][k1]
  }

  const float* __restrict__ xrow = c + (size_t)row * S;
  float* __restrict__ yrow = out + (size_t)row * S;

  const int tile_lane_off = 16 * r + c0;  // A-operand base offset within a tile
  const int n_tiles = S >> 8;             // 256 elements per tile
  float carry = 0.0f;

  for (int t = 0; t < n_tiles; ++t) {
    const float* tp = xrow + (t << 8);

    if (t + 1 < n_tiles) {
      __builtin_prefetch(tp + 256 + tile_lane_off, 0, 3);  // global_prefetch
    }

    // Load X in A-operand layout (4 x b64 per lane) and apply softplus.
    v2f A[4];
#pragma unroll
    for (int s = 0; s < 4; ++s) {
      v2f raw = *(const v2f*)(tp + tile_lane_off + 4 * s);
      A[s][0] = softplus_beta(raw[0], bl, lc);
      A[s][1] = softplus_beta(raw[1], bl, lc);
    }

    // D1 = X * U  (prefix sums within each 16-element tile row)
    v8f acc = {};
#pragma unroll
    for (int s = 0; s < 4; ++s) acc = wmma4(A[s], UB[s], acc);

    // Build Z (rowsum matrix) in B layout from D1's column 15 via ds_swizzle.
    // z[v]: lo = rowsum_v,   hi = rowsum_{v+8}
    // w[v]: lo = rowsum_{v+8}, hi = rowsum_v
    float z0 = swz<SWZ_BCAST>(acc[0]), z1 = swz<SWZ_BCAST>(acc[1]);
    float z2 = swz<SWZ_BCAST>(acc[2]), z3 = swz<SWZ_BCAST>(acc[3]);
    float z4 = swz<SWZ_BCAST>(acc[4]), z5 = swz<SWZ_BCAST>(acc[5]);
    float z6 = swz<SWZ_BCAST>(acc[6]), z7 = swz<SWZ_BCAST>(acc[7]);
    float w0 = swz<SWZ_CROSS>(acc[0]), w1 = swz<SWZ_CROSS>(acc[1]);
    float w2 = swz<SWZ_CROSS>(acc[2]), w3 = swz<SWZ_CROSS>(acc[3]);
    float w4 = swz<SWZ_CROSS>(acc[4]), w5 = swz<SWZ_CROSS>(acc[5]);
    float w6 = swz<SWZ_CROSS>(acc[6]), w7 = swz<SWZ_CROSS>(acc[7]);

    // B-operand slices of Z: slice s holds Z rows 4s..4s+3.
    v2f ZB[4];
    ZB[0][0] = hi ? w2 : z0;  ZB[0][1] = hi ? w3 : z1;  // rows 0,1 | 2,3
    ZB[1][0] = hi ? w6 : z4;  ZB[1][1] = hi ? w7 : z5;  // rows 4,5 | 6,7
    ZB[2][0] = hi ? z2 : w0;  ZB[2][1] = hi ? z3 : w1;  // rows 8,9 | 10,11
    ZB[3][0] = hi ? z6 : w4;  ZB[3][1] = hi ? z7 : w5;  // rows 12,13 | 14,15

    // D = D1 + Lstrict * Z  (accumulate straight onto D1)
#pragma unroll
    for (int s = 0; s < 4; ++s) acc = wmma4(LA[s], ZB[s], acc);

    // Add the running carry, store (coalesced b32, 2x64B per instruction),
    // and pick up the new carry from the tile's last element (lane 31, VGPR 7).
    float* op = yrow + (t << 8) + (hi ? 112 : 0) + lane;
    float last = 0.0f;
#pragma unroll
    for (int v = 0; v < 8; ++v) {
      float val = acc[v] + carry;
      op[16 * v] = val;
      if (v == 7) last = val;
    }
    carry = __int_as_float(__builtin_amdgcn_readlane(__float_as_int(last), 31));
  }
}

extern "C" void kernel_launch(void* const* d_in, const int* in_sizes, int n_in,
                              void* d_out, int out_size, void* d_ws, size_t ws_size,
                              hipStream_t stream) {
  const float* c = (const float*)d_in[0];
  const float* beta = (const float*)d_in[1];
  float* out = (float*)d_out;

  const int S = 8192;                 // row length (fixed by reference)
  const int B = in_sizes[0] / S;      // number of rows

  const int waves_per_block = 8;      // 256 threads = 8 wave32
  dim3 block(32 * waves_per_block);
  dim3 grid((B + waves_per_block - 1) / waves_per_block);
  softplus_cumsum_wmma<<<grid, block, 0, stream>>>(c, beta, out, B, S);
}